// VIEW1_9345848836754
// MI455X (gfx1250) — compile-verified
//
#include <hip/hip_runtime.h>
#include <hip/hip_bf16.h>
#include <math.h>

typedef __attribute__((ext_vector_type(2))) float v2f;
typedef __attribute__((ext_vector_type(8))) float v8f;

#define DIN  64
#define DOUT 64
#define NEG_SLOPE 0.2f

// ---------------------------------------------------------------------------
// float atomic max via integer punning (works for all signs; m init = -inf)
// ---------------------------------------------------------------------------
__device__ __forceinline__ void atomicMaxFloat(float* addr, float value) {
    if (value >= 0.0f) {
        atomicMax((int*)addr, __float_as_int(value));
    } else {
        atomicMin((unsigned int*)addr, __float_as_uint(value));
    }
}

// ---------------------------------------------------------------------------
// out[n][c] = b_concur[c] + b_upd[c] + b_side[c]   (also clears poison)
// ---------------------------------------------------------------------------
__global__ void init_out_kernel(float* __restrict__ out,
                                const float* __restrict__ b0,
                                const float* __restrict__ b1,
                                const float* __restrict__ b2,
                                int total) {
    int i = blockIdx.x * blockDim.x + threadIdx.x;
    if (i >= total) return;
    int c = i & (DOUT - 1);
    out[i] = b0[c] + b1[c] + b2[c];
}

__global__ void init_ms_kernel(float* __restrict__ m, float* __restrict__ s, int n) {
    int i = blockIdx.x * blockDim.x + threadIdx.x;
    if (i >= n) return;
    m[i] = -INFINITY;
    s[i] = 0.0f;
}

// ---------------------------------------------------------------------------
// feat = x @ W  via V_WMMA_F32_16X16X4_F32 (fp32 WMMA, exact precision)
// block = 256 threads = 8 waves; wave w -> rowTile (w>>2) in {0,1},
// colTile (w&3) in {0..3}; block covers 32 rows x 64 cols.
// A 16x4 f32: lanes0-15 M=lane, VGPR0=K0,VGPR1=K1; lanes16-31 K2,K3.
// B 4x16 f32: VGPR0 lanes0-15 K=0 / lanes16-31 K=2; VGPR1 K=1 / K=3.
// C 16x16 f32: VGPR r -> (M=r lanes0-15 | M=r+8 lanes16-31), N=lane%16.
// EXEC stays all-ones through the WMMA loop (no predication; row clamped,
// stores masked afterwards).
// ---------------------------------------------------------------------------
__global__ void gat_gemm_feat(const float* __restrict__ x,
                              const float* __restrict__ W,
                              float* __restrict__ feat,
                              int n_nodes) {
    __shared__ float sW[DIN * DOUT];
    for (int i = threadIdx.x; i < DIN * DOUT; i += 256) sW[i] = W[i];
    __syncthreads();

    const int wave    = threadIdx.x >> 5;
    const int lane    = threadIdx.x & 31;
    const int rowTile = wave >> 2;   // 0..1
    const int colTile = wave & 3;    // 0..3
    const int m0 = blockIdx.x * 32 + rowTile * 16;
    const int n0 = colTile * 16;

    const int half = lane >> 4;      // 0: lanes 0-15, 1: lanes 16-31
    const int l16  = lane & 15;

    // clamp the A row (no predication inside the loop; EXEC stays full)
    int arow = m0 + l16;
    if (arow >= n_nodes) arow = n_nodes - 1;
    const float2* __restrict__ xrow2 =
        (const float2*)(x + (size_t)arow * DIN);

    // preload B fragments (depend only on colTile/lane): 16 x v2f registers
    v2f bfrag[16];
    #pragma unroll
    for (int t = 0; t < 16; ++t) {
        const int ka = t * 4 + half * 2;
        bfrag[t].x = sW[(ka)     * DOUT + n0 + l16];
        bfrag[t].y = sW[(ka + 1) * DOUT + n0 + l16];
    }

    v8f acc = {};
    #pragma unroll
    for (int t = 0; t < 16; ++t) {
        // element offset = (t*2 + half)*2 = t*4 + half*2  -> one b64 load
        const float2 av = xrow2[t * 2 + half];
        v2f a;
        a.x = av.x;
        a.y = av.y;
        acc = __builtin_amdgcn_wmma_f32_16x16x4_f32(
            /*neg_a=*/false, a, /*neg_b=*/false, bfrag[t],
            /*c_mod=*/(short)0, acc, /*reuse_a=*/false, /*reuse_b=*/false);
    }

    #pragma unroll
    for (int r = 0; r < 8; ++r) {
        const int row = m0 + r + half * 8;
        if (row < n_nodes) feat[(size_t)row * DOUT + n0 + l16] = acc[r];
    }
}

// ---------------------------------------------------------------------------
// el[n] = feat[n,:] . al ; er[n] = feat[n,:] . ar   (b128 feat loads)
// ---------------------------------------------------------------------------
__global__ void gat_node_logits(const float* __restrict__ feat,
                                const float* __restrict__ al,
                                const float* __restrict__ ar,
                                float* __restrict__ el,
                                float* __restrict__ er,
                                int n) {
    int node = blockIdx.x * blockDim.x + threadIdx.x;
    if (node >= n) return;
    const float4* f4 = (const float4*)(feat + (size_t)node * DOUT);
    float sl = 0.0f, sr = 0.0f;
    #pragma unroll
    for (int i = 0; i < DOUT / 4; ++i) {
        float4 v = f4[i];
        sl += v.x * al[i * 4]     + v.y * al[i * 4 + 1]
            + v.z * al[i * 4 + 2] + v.w * al[i * 4 + 3];
        sr += v.x * ar[i * 4]     + v.y * ar[i * 4 + 1]
            + v.z * ar[i * 4 + 2] + v.w * ar[i * 4 + 3];
    }
    el[node] = sl;
    er[node] = sr;
}

// ---------------------------------------------------------------------------
// Edge pass 1: e = leaky_relu(el[src]+er[dst]); store e; segment max into m
// ---------------------------------------------------------------------------
__global__ void edge_logits_max(const int* __restrict__ src,
                                const int* __restrict__ dst,
                                const float* __restrict__ el,
                                const float* __restrict__ er,
                                float* __restrict__ ebuf,
                                float* __restrict__ m,
                                int ne) {
    int e = blockIdx.x * blockDim.x + threadIdx.x;
    if (e >= ne) return;
    int d = dst[e];
    float t = el[src[e]] + er[d];
    t = (t > 0.0f) ? t : NEG_SLOPE * t;
    ebuf[e] = t;
    atomicMaxFloat(&m[d], t);
}

// ---------------------------------------------------------------------------
// Edge pass 2: s[dst] += exp(e - m[dst])
// ---------------------------------------------------------------------------
__global__ void edge_expsum(const int* __restrict__ dst,
                            const float* __restrict__ ebuf,
                            const float* __restrict__ m,
                            float* __restrict__ s,
                            int ne) {
    int e = blockIdx.x * blockDim.x + threadIdx.x;
    if (e >= ne) return;
    int d = dst[e];
    atomicAdd(&s[d], __expf(ebuf[e] - m[d]));
}

// ---------------------------------------------------------------------------
// Edge pass 3: out[dst] += (exp(e - m[dst]) / s[dst]) * feat[src]
// one wave32 per edge; each lane handles 2 output columns (one float2 load).
// per-edge scalars forced onto the scalar path via readfirstlane.
// ---------------------------------------------------------------------------
__global__ void edge_aggregate(const int* __restrict__ src,
                               const int* __restrict__ dst,
                               const float* __restrict__ ebuf,
                               const float* __restrict__ m,
                               const float* __restrict__ s,
                               const float* __restrict__ feat,
                               float* __restrict__ out,
                               int ne) {
    long long idx = (long long)blockIdx.x * blockDim.x + threadIdx.x;
    // idx>>5 is wave-uniform; make it provably scalar for the compiler
    int e    = __builtin_amdgcn_readfirstlane((int)(idx >> 5));
    int lane = (int)(idx & 31);
    if (e >= ne) return;
    int d  = dst[e];   // scalar loads (uniform address)
    int sc = src[e];
    float coef = __expf(ebuf[e] - m[d]) / s[d];
    const float2* f2 = (const float2*)(feat + (size_t)sc * DOUT);
    float*        o  = out + (size_t)d * DOUT;
    float2 fv = f2[lane];
    int c = lane * 2;
    atomicAdd(&o[c],     coef * fv.x);
    atomicAdd(&o[c + 1], coef * fv.y);
}

// ---------------------------------------------------------------------------
// Host-side orchestration
// ---------------------------------------------------------------------------
extern "C" void kernel_launch(void* const* d_in, const int* in_sizes, int n_in,
                              void* d_out, int out_size, void* d_ws, size_t ws_size,
                              hipStream_t stream) {
    const float* x = (const float*)d_in[0];
    const int N = in_sizes[0] / DIN;

    float* out = (float*)d_out;

    // workspace layout (all fp32)
    char* ws = (char*)d_ws;
    float* feat = (float*)ws; ws += (size_t)N * DOUT * sizeof(float);
    float* el   = (float*)ws; ws += (size_t)N * sizeof(float);
    float* er   = (float*)ws; ws += (size_t)N * sizeof(float);
    float* mbuf = (float*)ws; ws += (size_t)N * sizeof(float);
    float* sbuf = (float*)ws; ws += (size_t)N * sizeof(float);
    float* ebuf = (float*)ws;  // E floats, reused per relation

    // out = b_concur + b_upd + b_side (broadcast per column)
    {
        const float* b0 = (const float*)d_in[6];
        const float* b1 = (const float*)d_in[12];
        const float* b2 = (const float*)d_in[18];
        int total = N * DOUT;
        init_out_kernel<<<(total + 255) / 256, 256, 0, stream>>>(out, b0, b1, b2, total);
    }

    for (int r = 0; r < 3; ++r) {
        const int*   src = (const int*)d_in[1 + r * 6];
        const int*   dst = (const int*)d_in[2 + r * 6];
        const float* W   = (const float*)d_in[3 + r * 6];
        const float* al  = (const float*)d_in[4 + r * 6];
        const float* ar  = (const float*)d_in[5 + r * 6];
        const int    ne  = in_sizes[1 + r * 6];

        // feat = x @ W  (WMMA fp32), 32 rows per block
        gat_gemm_feat<<<(N + 31) / 32, 256, 0, stream>>>(x, W, feat, N);

        // el / er
        gat_node_logits<<<(N + 255) / 256, 256, 0, stream>>>(feat, al, ar, el, er, N);

        // softmax state
        init_ms_kernel<<<(N + 255) / 256, 256, 0, stream>>>(mbuf, sbuf, N);

        // edge softmax + aggregation
        edge_logits_max<<<(ne + 255) / 256, 256, 0, stream>>>(src, dst, el, er, ebuf, mbuf, ne);
        edge_expsum<<<(ne + 255) / 256, 256, 0, stream>>>(dst, ebuf, mbuf, sbuf, ne);

        long long tot = (long long)ne * 32;
        int blocks = (int)((tot + 255) / 256);
        edge_aggregate<<<blocks, 256, 0, stream>>>(src, dst, ebuf, mbuf, sbuf, feat, out, ne);
    }
}